// VideoQuantizer_7541962572023
// MI455X (gfx1250) — compile-verified
//
#include <hip/hip_runtime.h>

typedef __attribute__((ext_vector_type(2))) float v2f;
typedef __attribute__((ext_vector_type(8))) float v8f;

#define B_DIM 8
#define T_DIM 1024
#define D_DIM 1024
#define Q_DIM 8
#define K_CODES 4096
#define D_SUB 128            // d = D/Q
#define M_ROWS (B_DIM * T_DIM)

#if __has_builtin(__builtin_amdgcn_sched_barrier)
#define SCHED_FENCE() __builtin_amdgcn_sched_barrier(0)
#else
#define SCHED_FENCE() asm volatile("" ::: "memory")
#endif

// ---------------------------------------------------------------------------
// Kernel 1: input RMSNorm.  One block (256 thr) per row of 1024 fp32.
// ---------------------------------------------------------------------------
__global__ __launch_bounds__(256) void k_rmsnorm_in(const float* __restrict__ x,
                                                    const float* __restrict__ w,
                                                    float* __restrict__ xn) {
  __shared__ float sm[8];
  const int row = blockIdx.x;
  const float4* xr = (const float4*)(x + (size_t)row * D_DIM);
  float4 v = xr[threadIdx.x];
  float ss = v.x * v.x + v.y * v.y + v.z * v.z + v.w * v.w;
#pragma unroll
  for (int off = 16; off; off >>= 1) ss += __shfl_xor(ss, off, 32);
  if ((threadIdx.x & 31) == 0) sm[threadIdx.x >> 5] = ss;
  __syncthreads();
  if (threadIdx.x == 0) {
    float s = 0.f;
#pragma unroll
    for (int i = 0; i < 8; ++i) s += sm[i];
    sm[0] = s;
  }
  __syncthreads();
  const float scale = rsqrtf(sm[0] * (1.0f / D_DIM) + 1e-5f);
  float4 wv = ((const float4*)w)[threadIdx.x];
  float4 o;
  o.x = v.x * scale * wv.x;
  o.y = v.y * scale * wv.y;
  o.z = v.z * scale * wv.z;
  o.w = v.w * scale * wv.w;
  ((float4*)(xn + (size_t)row * D_DIM))[threadIdx.x] = o;
}

// ---------------------------------------------------------------------------
// Kernel 1b: precompute codebook squared norms c2[q*K + k] (128 KB table,
// L2-resident for the whole distance scan).  One thread per code.
// ---------------------------------------------------------------------------
__global__ __launch_bounds__(256) void k_c2(const float* __restrict__ cb,
                                            float* __restrict__ c2) {
  const int code = blockIdx.x * 256 + threadIdx.x;  // 0 .. Q*K-1
  const float4* p = (const float4*)(cb + (size_t)code * D_SUB);
  float s = 0.f;
#pragma unroll
  for (int i = 0; i < D_SUB / 4; ++i) {
    float4 v = p[i];
    s += v.x * v.x + v.y * v.y + v.z * v.z + v.w * v.w;
  }
  c2[code] = s;
}

// ---------------------------------------------------------------------------
// Kernel 2: WMMA distance computation + argmin over K codes.
// Grid: (M_ROWS/16, Q).  Block: 128 threads = 4 waves.
// Each wave keeps the 16x128 A panel in registers as 32 v2f fragments and
// scans 64 column tiles of 16 codes with v_wmma_f32_16x16x4_f32.  The whole
// B tile is staged in registers behind a sched_barrier so all 32 b64 loads
// are in flight before the WMMA chain consumes them.  Two accumulator
// chains (even/odd k) double the ILP of the matrix pipe.
// ---------------------------------------------------------------------------
__global__ __launch_bounds__(128)
__attribute__((amdgpu_waves_per_eu(1)))
void k_argmin(const float* __restrict__ xn,
              const float* __restrict__ cb,
              const float* __restrict__ c2tab,
              int* __restrict__ idx_out) {
  __shared__ float sbest[4][16];
  __shared__ int sidx[4][16];

  const int tile = blockIdx.x;     // 16-row tile of M
  const int q = blockIdx.y;
  const int lane = threadIdx.x & 31;
  const int wave = threadIdx.x >> 5;
  const int half = lane >> 4;      // 0: K sub 0,1   1: K sub 2,3 (x4 A/B layout)
  const int l16 = lane & 15;
  const int row0 = tile * 16;

  const float* Aptr = xn + (size_t)row0 * D_DIM + q * D_SUB;
  const float* Bq = cb + (size_t)q * K_CODES * D_SUB;
  const float* c2q = c2tab + (size_t)q * K_CODES;

  // --- load A fragments (16x4 per k-step, 32 k-steps cover d=128) ---------
  v2f a[32];
  float x2p = 0.f;
  {
    const float* ap = Aptr + (size_t)l16 * D_DIM + half * 2;
#pragma unroll
    for (int ks = 0; ks < 32; ++ks) {
      float2 t = *(const float2*)(ap + ks * 4);
      a[ks].x = t.x;
      a[ks].y = t.y;
      x2p += t.x * t.x + t.y * t.y;
    }
  }
  // lane l (both halves) now holds |x_row|^2 for row (l & 15)
  const float x2row = x2p + __shfl_xor(x2p, 16, 32);

  // Hoisted out of the scan loop: |x|^2 for the 8 C-rows this lane covers
  // (C/D layout: M = r + 8*(lane/16)).
  float x2m[8];
#pragma unroll
  for (int r = 0; r < 8; ++r) x2m[r] = __shfl(x2row, r + half * 8, 32);

  float best[8];
  int bidx[8];
#pragma unroll
  for (int r = 0; r < 8; ++r) {
    best[r] = 3.4e38f;
    bidx[r] = 0x7FFFFFFF;
  }

  // --- scan column tiles of 16 codes, interleaved across the 4 waves ------
  for (int nt = wave; nt < K_CODES / 16; nt += 4) {
    const int code = nt * 16 + l16;  // B-matrix N = lane & 15
    const float* bp = Bq + (size_t)code * D_SUB + half * 2;
    const float c2 = c2q[code];

    // Stage the full B tile in registers.  The sched_barrier below forbids
    // the machine scheduler from sinking these loads back to their WMMA
    // uses, so all 32 b64 loads issue first and stay in flight.
    v2f b[32];
#pragma unroll
    for (int ks = 0; ks < 32; ++ks) {
      float2 t = *(const float2*)(bp + ks * 4);
      b[ks].x = t.x;
      b[ks].y = t.y;
    }
    SCHED_FENCE();

    v8f acc0 = {};
    v8f acc1 = {};
#pragma unroll
    for (int ks = 0; ks < 32; ks += 2) {
      acc0 = __builtin_amdgcn_wmma_f32_16x16x4_f32(
          false, a[ks], false, b[ks], (short)0, acc0, false, false);
      acc1 = __builtin_amdgcn_wmma_f32_16x16x4_f32(
          false, a[ks + 1], false, b[ks + 1], (short)0, acc1, false, false);
    }

#pragma unroll
    for (int r = 0; r < 8; ++r) {
      const float xc = acc0[r] + acc1[r];
      const float dist = x2m[r] - 2.0f * xc + c2;
      if (dist < best[r] || (dist == best[r] && code < bidx[r])) {
        best[r] = dist;
        bidx[r] = code;
      }
    }
  }

  // --- reduce argmin over N (16 lanes per half) ---------------------------
#pragma unroll
  for (int r = 0; r < 8; ++r) {
    float bv = best[r];
    int bi = bidx[r];
#pragma unroll
    for (int off = 8; off; off >>= 1) {
      const float ov = __shfl_xor(bv, off, 32);
      const int oi = __shfl_xor(bi, off, 32);
      if (ov < bv || (ov == bv && oi < bi)) {
        bv = ov;
        bi = oi;
      }
    }
    if (l16 == 0) {
      sbest[wave][r + half * 8] = bv;
      sidx[wave][r + half * 8] = bi;
    }
  }
  __syncthreads();

  // --- reduce across the 4 waves, emit indices ----------------------------
  if (threadIdx.x < 16) {
    const int m = threadIdx.x;
    float bv = sbest[0][m];
    int bi = sidx[0][m];
#pragma unroll
    for (int w2 = 1; w2 < 4; ++w2) {
      const float ov = sbest[w2][m];
      const int oi = sidx[w2][m];
      if (ov < bv || (ov == bv && oi < bi)) {
        bv = ov;
        bi = oi;
      }
    }
    idx_out[(size_t)(row0 + m) * Q_DIM + q] = bi;
  }
}

// ---------------------------------------------------------------------------
// Kernel 3: gather selected code vectors + output RMSNorm.
// One block (256 thr) per (b,t) row.
// ---------------------------------------------------------------------------
__global__ __launch_bounds__(256) void k_output(const float* __restrict__ cb,
                                                const int* __restrict__ idx,
                                                const float* __restrict__ w,
                                                float* __restrict__ out) {
  __shared__ float sm[8];
  __shared__ int scode[Q_DIM];
  const int row = blockIdx.x;
  if (threadIdx.x < Q_DIM) scode[threadIdx.x] = idx[(size_t)row * Q_DIM + threadIdx.x];
  __syncthreads();

  const int e0 = threadIdx.x * 4;  // element offset within D=1024
  const int q = e0 >> 7;           // 128 elements per q-slot
  const int doff = e0 & 127;
  const int code = scode[q];
  float4 v = *(const float4*)(cb + ((size_t)q * K_CODES + code) * D_SUB + doff);

  float ss = v.x * v.x + v.y * v.y + v.z * v.z + v.w * v.w;
#pragma unroll
  for (int off = 16; off; off >>= 1) ss += __shfl_xor(ss, off, 32);
  if ((threadIdx.x & 31) == 0) sm[threadIdx.x >> 5] = ss;
  __syncthreads();
  if (threadIdx.x == 0) {
    float s = 0.f;
#pragma unroll
    for (int i = 0; i < 8; ++i) s += sm[i];
    sm[0] = s;
  }
  __syncthreads();
  const float scale = rsqrtf(sm[0] * (1.0f / D_DIM) + 1e-5f);
  float4 wv = ((const float4*)w)[threadIdx.x];
  float4 o;
  o.x = v.x * scale * wv.x;
  o.y = v.y * scale * wv.y;
  o.z = v.z * scale * wv.z;
  o.w = v.w * scale * wv.w;
  ((float4*)(out + (size_t)row * D_DIM))[threadIdx.x] = o;
}

// ---------------------------------------------------------------------------
extern "C" void kernel_launch(void* const* d_in, const int* in_sizes, int n_in,
                              void* d_out, int out_size, void* d_ws, size_t ws_size,
                              hipStream_t stream) {
  (void)in_sizes;
  (void)n_in;
  (void)out_size;
  (void)ws_size;
  const float* x = (const float*)d_in[0];
  const float* cb = (const float*)d_in[1];
  const float* w_in = (const float*)d_in[2];
  const float* w_out = (const float*)d_in[3];

  float* out = (float*)d_out;
  // tuple output: [out (B*T*D f32)] ++ [idx (B*T*Q i32)]
  int* idx_out = (int*)(out + (size_t)M_ROWS * D_DIM);

  // workspace layout: xn (M*D f32 = 32 MB) ++ c2 table (Q*K f32 = 128 KB)
  float* xn = (float*)d_ws;
  float* c2 = xn + (size_t)M_ROWS * D_DIM;

  k_rmsnorm_in<<<M_ROWS, 256, 0, stream>>>(x, w_in, xn);
  k_c2<<<(Q_DIM * K_CODES) / 256, 256, 0, stream>>>(cb, c2);

  dim3 g2(M_ROWS / 16, Q_DIM);
  k_argmin<<<g2, 128, 0, stream>>>(xn, cb, c2, idx_out);

  k_output<<<M_ROWS, 256, 0, stream>>>(cb, idx_out, w_out, out);
}